// three_gcn_1460288880956
// MI455X (gfx1250) — compile-verified
//
#include <hip/hip_runtime.h>

// ---------------------------------------------------------------------------
// 3-layer PyG GraphConv (aggr='add') for MI455X (gfx1250, wave32).
//   per layer: agg = segment_sum(x[src], dst); y = act(agg@W_rel + b + x@W_root)
// Memory-bound (~2 GB traffic -> ~90us floor @ 23.3 TB/s). GEMMs are ~5 GFLOP
// total, so we keep them in exact fp32 using V_WMMA_F32_16X16X4_F32.
// ---------------------------------------------------------------------------

typedef float v2f __attribute__((ext_vector_type(2)));
typedef float v8f __attribute__((ext_vector_type(8)));

#define NODES 100000
#define EDGES 1250000
#define D     64
#define TILE_ROWS (NODES / 16)      // 6250
#define TILE_COLS (D / 16)          // 4
#define N_TILES   (TILE_ROWS * TILE_COLS)  // 25000
#define WAVES_PER_BLOCK 8           // 256 threads = 8 wave32

// --------------------------- zero workspace --------------------------------
__global__ void zero_f32_kernel(float* __restrict__ p, int n) {
  int i = blockIdx.x * blockDim.x + threadIdx.x;
  if (i < n) p[i] = 0.0f;
}

// ------------------------ edge scatter-add (agg) ---------------------------
// One thread per (edge, 16-float chunk): E*4 = 5M threads.
// float4 gathers from x[src]; fp32 atomic adds into agg[dst] (L2-resident).
__global__ void scatter_add_kernel(const float* __restrict__ x,
                                   const int* __restrict__ src,
                                   const int* __restrict__ dst,
                                   float* __restrict__ agg) {
  int idx = blockIdx.x * blockDim.x + threadIdx.x;
  if (idx >= EDGES * 4) return;
  int e = idx >> 2;
  int q = (idx & 3) * 16;                 // feature chunk base (16 floats)
  int s = src[e];
  int d = dst[e];
  const float4* xs = (const float4*)(x + (size_t)s * D + q);
  float* ad = agg + (size_t)d * D + q;
#pragma unroll
  for (int j = 0; j < 4; ++j) {
    float4 v = xs[j];
    unsafeAtomicAdd(ad + 4 * j + 0, v.x);
    unsafeAtomicAdd(ad + 4 * j + 1, v.y);
    unsafeAtomicAdd(ad + 4 * j + 2, v.z);
    unsafeAtomicAdd(ad + 4 * j + 3, v.w);
  }
}

// ---------------- WMMA GEMM: out = act(agg@Wrel + b + x@Wroot) -------------
// One wave32 per 16x16 output tile. Exact fp32 via V_WMMA_F32_16X16X4_F32.
//
// A (16x4 f32) layout: lanes 0-15 -> rows M=0..15 holding K=0 (vgpr0), K=1
// (vgpr1); lanes 16-31 -> same rows holding K=2,3.
// B (4x16 f32) layout: lanes 0-15 -> cols N=0..15 of rows K=0 (vgpr0), K=1
// (vgpr1); lanes 16-31 -> rows K=2,3.
// C/D (16x16 f32): vgpr r -> M=r (lanes 0-15) / M=r+8 (lanes 16-31), N=lane%16.
__global__ void __launch_bounds__(256)
gconv_wmma_kernel(const float* __restrict__ agg,
                  const float* __restrict__ x,
                  const float* __restrict__ Wrel,
                  const float* __restrict__ Wroot,
                  const float* __restrict__ bias,
                  float* __restrict__ out,
                  int act /* 0 = ELU, 1 = sigmoid */) {
  const int wave = threadIdx.x >> 5;
  const int lane = threadIdx.x & 31;
  const int tile = blockIdx.x * WAVES_PER_BLOCK + wave;   // wave-uniform
  if (tile >= N_TILES) return;                            // uniform branch

  const int trow = tile / TILE_COLS;       // node-tile index
  const int tcol = tile % TILE_COLS;       // feature-tile index
  const int lg   = lane >> 4;              // lane group: 0 or 1
  const int lm   = lane & 15;
  const int arow = trow * 16 + lm;         // A row this lane reads
  const int col  = tcol * 16 + lm;         // B/C column this lane owns
  const int koff = lg * 2;                 // K sub-offset per lane group

  const float* __restrict__ aA = agg + (size_t)arow * D;
  const float* __restrict__ xA = x   + (size_t)arow * D;

  v8f c = {};  // fp32 accumulator, starts at zero

#pragma unroll
  for (int k = 0; k < D; k += 4) {        // agg @ W_rel
    v2f a, b;
    a.x = aA[k + koff];
    a.y = aA[k + koff + 1];
    b.x = Wrel[(k + koff) * D + col];
    b.y = Wrel[(k + koff + 1) * D + col];
    c = __builtin_amdgcn_wmma_f32_16x16x4_f32(false, a, false, b,
                                              (short)0, c, false, false);
  }
#pragma unroll
  for (int k = 0; k < D; k += 4) {        // x @ W_root
    v2f a, b;
    a.x = xA[k + koff];
    a.y = xA[k + koff + 1];
    b.x = Wroot[(k + koff) * D + col];
    b.y = Wroot[(k + koff + 1) * D + col];
    c = __builtin_amdgcn_wmma_f32_16x16x4_f32(false, a, false, b,
                                              (short)0, c, false, false);
  }

  const float bb = bias[col];
#pragma unroll
  for (int r = 0; r < 8; ++r) {
    const int orow = trow * 16 + r + lg * 8;   // C/D layout: M = r or r+8
    float v = c[r] + bb;
    float y;
    if (act == 0) {
      y = (v > 0.0f) ? v : expm1f(v);          // ELU(alpha=1)
    } else {
      y = 1.0f / (1.0f + expf(-v));            // sigmoid
    }
    out[(size_t)orow * D + col] = y;
  }
}

// ---------------------------------------------------------------------------
extern "C" void kernel_launch(void* const* d_in, const int* in_sizes, int n_in,
                              void* d_out, int out_size, void* d_ws, size_t ws_size,
                              hipStream_t stream) {
  const float* graph  = (const float*)d_in[0];
  const int*   edges  = (const int*)d_in[1];     // [2, E]
  const float* W1_rel = (const float*)d_in[2];
  const float* W1_rt  = (const float*)d_in[3];
  const float* b1     = (const float*)d_in[4];
  const float* W2_rel = (const float*)d_in[5];
  const float* W2_rt  = (const float*)d_in[6];
  const float* b2     = (const float*)d_in[7];
  const float* W3_rel = (const float*)d_in[8];
  const float* W3_rt  = (const float*)d_in[9];
  const float* b3     = (const float*)d_in[10];

  const int* src = edges;          // edge_index[0]
  const int* dst = edges + EDGES;  // edge_index[1]

  float* agg = (float*)d_ws;                       // N*D fp32 = 25.6 MB
  float* y1  = (float*)d_out;                      // [N, D]
  float* y2  = y1 + (size_t)NODES * D;
  float* y3  = y2 + (size_t)NODES * D;

  const int nd = NODES * D;
  const dim3 blk(256);
  const dim3 grid_zero((nd + 255) / 256);
  const dim3 grid_scat((EDGES * 4 + 255) / 256);
  const dim3 grid_gemm((N_TILES + WAVES_PER_BLOCK - 1) / WAVES_PER_BLOCK);

  // ---- layer 1: y1 = elu(seg_sum(graph) @ W1_rel + b1 + graph @ W1_root)
  zero_f32_kernel<<<grid_zero, blk, 0, stream>>>(agg, nd);
  scatter_add_kernel<<<grid_scat, blk, 0, stream>>>(graph, src, dst, agg);
  gconv_wmma_kernel<<<grid_gemm, blk, 0, stream>>>(agg, graph, W1_rel, W1_rt, b1, y1, 0);

  // ---- layer 2: y2 = elu(seg_sum(y1) @ W2_rel + b2 + y1 @ W2_root)
  zero_f32_kernel<<<grid_zero, blk, 0, stream>>>(agg, nd);
  scatter_add_kernel<<<grid_scat, blk, 0, stream>>>(y1, src, dst, agg);
  gconv_wmma_kernel<<<grid_gemm, blk, 0, stream>>>(agg, y1, W2_rel, W2_rt, b2, y2, 0);

  // ---- layer 3: y3 = sigmoid(seg_sum(y2) @ W3_rel + b3 + y2 @ W3_root)
  zero_f32_kernel<<<grid_zero, blk, 0, stream>>>(agg, nd);
  scatter_add_kernel<<<grid_scat, blk, 0, stream>>>(y2, src, dst, agg);
  gconv_wmma_kernel<<<grid_gemm, blk, 0, stream>>>(agg, y2, W3_rel, W3_rt, b3, y3, 1);
}